// BidPrefix_83081847374046
// MI455X (gfx1250) — compile-verified
//
#include <hip/hip_runtime.h>
#include <hip/hip_bf16.h>
#include <stdint.h>

// BidPrefix: per-row prefix product gathered at 3 indices.
// Bandwidth-bound (~242 MB in, 2.4 MB out; HBM floor ~10.4us, ~7us with
// early exit). Strategy:
//   - one wave (32 lanes) per row
//   - early-exit at L = max(bid, mp+1)  (~33% traffic saved on average)
//   - CDNA5 async path: global_load_async_to_lds_b64 staging (coalesced
//     256B/issue, per-lane predicated, non-temporal), s_wait_asynccnt sync
//   - conflict-free ds_load_b128 consumption: each wave read is a
//     contiguous 512B LDS slice (byte stride 16 across lanes)
//   - masked per-element products + wave32 shfl_xor product reductions

#define SEQ_LEN    300
#define ROW_ELEMS  302
#define WAVES_PB   8

__global__ __launch_bounds__(256) void bidprefix_kernel(
    const float* __restrict__ in, float* __restrict__ out, int batch)
{
    __shared__ float smem[WAVES_PB][384];   // halves 0..2 cover elems 0..383

    const int lane  = threadIdx.x & 31;
    const int wslot = threadIdx.x >> 5;
    const int row   = blockIdx.x * WAVES_PB + wslot;
    if (row >= batch) return;

    const unsigned rowbytes = (unsigned)row * (unsigned)(ROW_ELEMS * 4);

    // Tail: bid (elem 300), mp (elem 301). rowbytes % 8 == 0 -> float2 OK.
    const float2 tail =
        *(const float2*)((const char*)in + rowbytes + SEQ_LEN * 4);
    const int bid = (int)tail.x;                       // 0..300
    const int mp  = (int)tail.y;                       // 0..299
    const int L   = (bid > mp + 1) ? bid : (mp + 1);   // 1..300

    // Raw LDS byte offset of this wave's buffer (flat addr low 32 bits).
    const unsigned wbase = (unsigned)(uintptr_t)&smem[wslot][0];

    // ---- Async stage: each issue moves a contiguous 256B slice (32 lanes
    // x b64), predicated per lane at e < L. Only s_wait_asynccnt 0 is used,
    // so EXEC==0 skip ambiguity on fully-predicated issues is harmless.
    #pragma unroll
    for (int j = 0; j < 5; ++j) {             // elems 0..319 covers L<=300
        const int e0 = 64 * j;
        if (e0 < L) {
            const int e = e0 + 2 * lane;      // this lane's pair of elems
            if (e < L) {
                unsigned loff = wbase + (unsigned)(e * 4);
                unsigned goff = rowbytes + (unsigned)(e * 4);
                asm volatile(
                    "global_load_async_to_lds_b64 %0, %1, %2 th:TH_LOAD_NT"
                    :: "v"(loff), "v"(goff), "s"(in)
                    : "memory");
            }
        }
    }
    asm volatile("s_wait_asynccnt 0x0" ::: "memory");

    // ---- Consume: 3 halves of 128 elems; each ds_load_b128 is a
    // contiguous 512B wave read (bank-conflict-free). Ownership pattern is
    // irrelevant to the commutative masked products.
    float acB = 1.0f;   // prod over idx < bid   -> cpz[bid]
    float acM = 1.0f;   // prod over idx < mp    -> cpz[mp]
    float rmp = 1.0f;   // rates[mp]
    const float4* lsm = (const float4*)&smem[wslot][0];

    #pragma unroll
    for (int h = 0; h < 3; ++h) {
        const int base = 128 * h;
        if (base < L) {                        // wave-uniform guard
            const int i0 = base + 4 * lane;    // lane's 4 consecutive elems
            const float4 a = lsm[32 * h + lane];          // ds_load_b128
            const float v[4] = { a.x, a.y, a.z, a.w };
            #pragma unroll
            for (int j = 0; j < 4; ++j) {
                const int   idx = i0 + j;
                const float r   = v[j];        // garbage if idx>=L: masked
                acB *= (idx < bid) ? r : 1.0f;
                acM *= (idx < mp ) ? r : 1.0f;
                rmp  = (idx == mp) ? r : rmp;
            }
        }
    }

    // ---- wave32 product reductions (non-contributing lanes hold 1.0)
    #pragma unroll
    for (int off = 16; off > 0; off >>= 1) {
        acB *= __shfl_xor(acB, off, 32);
        acM *= __shfl_xor(acM, off, 32);
        rmp *= __shfl_xor(rmp, off, 32);
    }

    if (lane == 0) {
        float* o = out + (size_t)row * 3;
        o[0] = acB;         // cpz[bid]
        o[1] = acM * rmp;   // cpz[mp+1] = cpz[mp] * rates[mp]
        o[2] = acM;         // cpz[mp]
    }
}

extern "C" void kernel_launch(void* const* d_in, const int* in_sizes, int n_in,
                              void* d_out, int out_size, void* d_ws, size_t ws_size,
                              hipStream_t stream) {
    const float* in  = (const float*)d_in[0];
    float*       out = (float*)d_out;
    const int batch  = in_sizes[0] / ROW_ELEMS;           // 200000
    const int blocks = (batch + WAVES_PB - 1) / WAVES_PB; // 25000
    hipLaunchKernelGGL(bidprefix_kernel, dim3(blocks), dim3(256), 0, stream,
                       in, out, batch);
}